// IterNorm_Single_29222957482780
// MI455X (gfx1250) — compile-verified
//
#include <hip/hip_runtime.h>

// ---- problem constants (match reference: X is (32, 64, 8192) fp32) ----
#define B_   32
#define C_   64
#define L_   8192
#define MTOT (B_ * L_)          // 262144 samples per channel
#define EPS_ 1e-05f

// kernel-1 staging: 256 samples per chunk, row stride padded to 260 words
#define S1      256
#define SW1     (S1 + 4)
#define NCHUNK  4               // chunks per block (register-accumulated)
#define NBLK1   ((B_ * (L_ / S1)) / NCHUNK)   // 256 blocks
// kernel-3 staging
#define S3   256
#define SW3  (S3 + 4)
// 64x64 matrices in kernel 2 padded to stride 68
#define SWM  68

typedef __attribute__((ext_vector_type(2))) float v2f;
typedef __attribute__((ext_vector_type(8))) float v8f;

// ---------------------------------------------------------------------------
// Kernel 1: Gram = sum_k x[i,k]*x[j,k]  (64x64), sums[c] = sum_k x[c,k].
// 256 blocks x 512 threads; each block grid-strides over 4 chunks of 256
// samples, accumulating its 16x16 WMMA tile in registers, then does ONE
// atomic pass at the end (256 atomics per gram element device-wide).
// ---------------------------------------------------------------------------
__global__ void __launch_bounds__(512)
iternorm_gram_kernel(const float* __restrict__ X,
                     float* __restrict__ gram,
                     float* __restrict__ sums)
{
    extern __shared__ float smem[];
    float* tile = smem;              // 64 x SW1
    float* lsum = smem + C_ * SW1;   // 64

    const int tid = threadIdx.x;
    if (tid < C_) lsum[tid] = 0.0f;

    const int wave = tid >> 5;
    const int lane = tid & 31;
    const int it = wave >> 2, jt = wave & 3;
    const int ml = lane & 15;
    const int kb = (lane >> 4) << 1;                // 0 or 2
    const float* arow = tile + (it * 16 + ml) * SW1;
    const float* brow = tile + (jt * 16 + ml) * SW1;

    // per-thread strip for channel sums: 8 threads per channel
    const int sc   = tid & 63;
    const int part = tid >> 6;                      // 0..7
    const float* srow = tile + sc * SW1 + part * (S1 / 8);

    v8f acc0 = {};
    v8f acc1 = {};
    float ssum = 0.0f;

    for (int ch = 0; ch < NCHUNK; ++ch) {
        const int chunk = blockIdx.x + ch * NBLK1;          // 0..1023
        const int b  = chunk / (L_ / S1);
        const int l0 = (chunk % (L_ / S1)) * S1;
        const float* Xb = X + (size_t)b * C_ * L_ + l0;

        __syncthreads();   // protect LDS re-staging vs. previous chunk's readers

        // stage 64x256 slab, float4 coalesced global -> LDS
        for (int idx = tid; idx < C_ * (S1 / 4); idx += 512) {
            const int c  = idx / (S1 / 4);
            const int l4 = (idx % (S1 / 4)) * 4;
            const float4 v = *reinterpret_cast<const float4*>(Xb + (size_t)c * L_ + l4);
            float* dst = tile + c * SW1 + l4;
            dst[0] = v.x; dst[1] = v.y; dst[2] = v.z; dst[3] = v.w;
        }
        __syncthreads();

        // channel-sum strip (register accumulate)
        #pragma unroll
        for (int i = 0; i < S1 / 8; ++i) ssum += srow[i];

        // WMMA: this wave's 16x16 tile, K = 256 in steps of 8 (dual chains)
        for (int k0 = 0; k0 < S1; k0 += 8) {
            v2f a0 = *reinterpret_cast<const v2f*>(arow + k0 + kb);
            v2f b0 = *reinterpret_cast<const v2f*>(brow + k0 + kb);
            acc0 = __builtin_amdgcn_wmma_f32_16x16x4_f32(false, a0, false, b0,
                                                         (short)0, acc0, false, false);
            v2f a1 = *reinterpret_cast<const v2f*>(arow + k0 + 4 + kb);
            v2f b1 = *reinterpret_cast<const v2f*>(brow + k0 + 4 + kb);
            acc1 = __builtin_amdgcn_wmma_f32_16x16x4_f32(false, a1, false, b1,
                                                         (short)0, acc1, false, false);
        }
    }

    // one atomic pass per block
    const int row0 = it * 16 + ((lane >> 4) << 3);  // +8 for upper half-wave
    const int col  = jt * 16 + ml;
    #pragma unroll
    for (int r = 0; r < 8; ++r)
        atomicAdd(&gram[(row0 + r) * C_ + col], acc0[r] + acc1[r]);

    __syncthreads();
    atomicAdd(&lsum[sc], ssum);
    __syncthreads();
    if (tid < C_) atomicAdd(&sums[tid], lsum[tid]);
}

// ---------------------------------------------------------------------------
// Kernel 2: single block. Sigma_N -> 5 Newton-Schulz iterations -> wm, mean.
// 16 waves, each owns one 16x16 tile of every 64x64 matmul.
// ---------------------------------------------------------------------------
__device__ __forceinline__ v8f mm_tile64(const float* A, const float* Bm,
                                         int it, int jt, int lane)
{
    const int ml = lane & 15;
    const int kb = (lane >> 4) << 1;
    const float* arow = A + (it * 16 + ml) * SWM;
    const float* bcol = Bm + jt * 16 + ml;
    v8f acc = {};
    #pragma unroll
    for (int k0 = 0; k0 < 64; k0 += 4) {
        v2f a = *reinterpret_cast<const v2f*>(arow + k0 + kb);
        v2f bb;
        bb.x = bcol[(k0 + kb) * SWM];
        bb.y = bcol[(k0 + kb + 1) * SWM];
        acc = __builtin_amdgcn_wmma_f32_16x16x4_f32(false, a, false, bb,
                                                    (short)0, acc, false, false);
    }
    return acc;
}

__global__ void __launch_bounds__(512)
iternorm_newton_kernel(const float* __restrict__ gram,
                       const float* __restrict__ sums,
                       float* __restrict__ wm_out,
                       float* __restrict__ mean_out)
{
    extern __shared__ float smem[];
    float* Smat = smem;                 // Sigma_N 64xSWM
    float* Pmat = Smat + 64 * SWM;
    float* Ta   = Pmat + 64 * SWM;
    float* Tb   = Ta   + 64 * SWM;
    float* mean = Tb   + 64 * SWM;      // 64
    float* sc   = mean + 64;            // [0]=trace

    const int tid = threadIdx.x;
    if (tid == 0) sc[0] = 0.0f;
    if (tid < 64) mean[tid] = sums[tid] * (1.0f / (float)MTOT);
    __syncthreads();

    float tloc = 0.0f;
    for (int idx = tid; idx < 64 * 64; idx += 512) {
        const int r = idx >> 6, c = idx & 63;
        float sig = gram[idx] * (1.0f / (float)MTOT) - mean[r] * mean[c];
        if (r == c) { sig += EPS_; tloc += sig; }
        Smat[r * SWM + c] = sig;
    }
    atomicAdd(&sc[0], tloc);
    __syncthreads();

    const float rTr = 1.0f / sc[0];
    for (int idx = tid; idx < 64 * 64; idx += 512) {
        const int r = idx >> 6, c = idx & 63;
        Smat[r * SWM + c] *= rTr;
        Pmat[r * SWM + c] = (r == c) ? 1.0f : 0.0f;
    }
    __syncthreads();

    const int wave = tid >> 5, lane = tid & 31;
    const int it = wave >> 2, jt = wave & 3;
    const int row0 = it * 16 + ((lane >> 4) << 3);
    const int col  = jt * 16 + (lane & 15);

    for (int t = 0; t < 5; ++t) {
        v8f acc = mm_tile64(Pmat, Pmat, it, jt, lane);      // Ta = P @ P
        #pragma unroll
        for (int r = 0; r < 8; ++r) Ta[(row0 + r) * SWM + col] = acc[r];
        __syncthreads();

        acc = mm_tile64(Ta, Pmat, it, jt, lane);            // Tb = Ta @ P
        #pragma unroll
        for (int r = 0; r < 8; ++r) Tb[(row0 + r) * SWM + col] = acc[r];
        __syncthreads();

        acc = mm_tile64(Tb, Smat, it, jt, lane);            // P = 1.5P - 0.5*Tb@S
        #pragma unroll
        for (int r = 0; r < 8; ++r) {
            const float p = Pmat[(row0 + r) * SWM + col];
            Pmat[(row0 + r) * SWM + col] = 1.5f * p - 0.5f * acc[r];
        }
        __syncthreads();
    }

    const float srt = sqrtf(rTr);
    for (int idx = tid; idx < 64 * 64; idx += 512) {
        const int r = idx >> 6, c = idx & 63;
        wm_out[idx] = Pmat[r * SWM + c] * srt;
    }
    if (tid < 64) mean_out[tid] = mean[tid];
}

// ---------------------------------------------------------------------------
// Kernel 3: Xn = wm @ (x - mean), 64x64 times 64x262144.
// ---------------------------------------------------------------------------
__global__ void __launch_bounds__(512)
iternorm_apply_kernel(const float* __restrict__ X,
                      const float* __restrict__ wm,
                      const float* __restrict__ mean,
                      float* __restrict__ out)
{
    extern __shared__ float smem[];
    float* xt    = smem;                 // 64 x SW3 (mean-subtracted)
    float* wmlds = xt + C_ * SW3;        // 64 x SWM
    float* mloc  = wmlds + 64 * SWM;     // 64

    const int tid = threadIdx.x;
    const int b   = blockIdx.x / (L_ / S3);
    const int l0  = (blockIdx.x % (L_ / S3)) * S3;
    const float* Xb = X + (size_t)b * C_ * L_ + l0;
    float*       Ob = out + (size_t)b * C_ * L_ + l0;

    if (tid < 64) mloc[tid] = mean[tid];
    for (int idx = tid; idx < 64 * 64; idx += 512) {
        const int r = idx >> 6, c = idx & 63;
        wmlds[r * SWM + c] = wm[idx];
    }
    __syncthreads();

    for (int idx = tid; idx < C_ * (S3 / 4); idx += 512) {
        const int c  = idx / (S3 / 4);
        const int l4 = (idx % (S3 / 4)) * 4;
        const float4 v = *reinterpret_cast<const float4*>(Xb + (size_t)c * L_ + l4);
        const float mc = mloc[c];
        float* dst = xt + c * SW3 + l4;
        dst[0] = v.x - mc; dst[1] = v.y - mc; dst[2] = v.z - mc; dst[3] = v.w - mc;
    }
    __syncthreads();

    const int wave = tid >> 5, lane = tid & 31;
    const int ml = lane & 15;
    const int kb = (lane >> 4) << 1;

    // 4 channel-tiles x 16 sample-tiles = 64 tiles; each wave does 4
    for (int tIdx = wave; tIdx < 4 * (S3 / 16); tIdx += 16) {
        const int it = tIdx & 3;
        const int nt = tIdx >> 2;
        const float* arow = wmlds + (it * 16 + ml) * SWM;
        const float* bcol = xt + nt * 16 + ml;
        v8f acc0 = {};
        v8f acc1 = {};
        #pragma unroll
        for (int k0 = 0; k0 < 64; k0 += 8) {
            v2f a0 = *reinterpret_cast<const v2f*>(arow + k0 + kb);
            v2f b0; b0.x = bcol[(k0 + kb) * SW3];
                    b0.y = bcol[(k0 + kb + 1) * SW3];
            acc0 = __builtin_amdgcn_wmma_f32_16x16x4_f32(false, a0, false, b0,
                                                         (short)0, acc0, false, false);
            v2f a1 = *reinterpret_cast<const v2f*>(arow + k0 + 4 + kb);
            v2f b1; b1.x = bcol[(k0 + 4 + kb) * SW3];
                    b1.y = bcol[(k0 + 4 + kb + 1) * SW3];
            acc1 = __builtin_amdgcn_wmma_f32_16x16x4_f32(false, a1, false, b1,
                                                         (short)0, acc1, false, false);
        }
        const int row0 = it * 16 + ((lane >> 4) << 3);
        const int col  = nt * 16 + ml;
        #pragma unroll
        for (int r = 0; r < 8; ++r)
            Ob[(size_t)(row0 + r) * L_ + col] = acc0[r] + acc1[r];
    }
}

// ---------------------------------------------------------------------------
extern "C" void kernel_launch(void* const* d_in, const int* in_sizes, int n_in,
                              void* d_out, int out_size, void* d_ws, size_t ws_size,
                              hipStream_t stream)
{
    const float* X   = (const float*)d_in[0];
    float*       out = (float*)d_out;
    float*       ws  = (float*)d_ws;

    float* gram = ws;          // 4096
    float* sums = ws + 4096;   // 64
    float* wm   = ws + 4160;   // 4096
    float* mean = ws + 8256;   // 64

    // zero the accumulators (Gram + sums) every call
    hipMemsetAsync(ws, 0, (4096 + 64) * sizeof(float), stream);

    const size_t lds1 = (size_t)(C_ * SW1 + 64) * sizeof(float);
    iternorm_gram_kernel<<<NBLK1, 512, lds1, stream>>>(X, gram, sums);

    const size_t lds2 = (size_t)(4 * 64 * SWM + 64 + 8) * sizeof(float);
    iternorm_newton_kernel<<<1, 512, lds2, stream>>>(gram, sums, wm, mean);

    const int nblk3 = B_ * (L_ / S3);   // 1024
    const size_t lds3 = (size_t)(C_ * SW3 + 64 * SWM + 64) * sizeof(float);
    iternorm_apply_kernel<<<nblk3, 512, lds3, stream>>>(X, wm, mean, out);
}